// Model_68186900792112
// MI455X (gfx1250) — compile-verified
//
#include <hip/hip_runtime.h>
#include <stdint.h>

// ---------------------------------------------------------------------------
// MLA KV-cache gather:  rows = kv_buffer[loc];  out = [k_nope | k_rope]
//   kv_buffer : (262144, 576) f32     loc : (65536,) i64
//   out       : k_nope (65536,512) f32 flat, then k_rope (65536,64) f32 flat
// Pure data movement: ~302 MB @ 23.3 TB/s => ~13 us floor. Zero FLOPs, so
// WMMA is inapplicable; the CDNA5 path that matters is the async
// global->LDS->global DMA engine (ASYNCcnt-tracked b128, no VGPR round-trip).
// ---------------------------------------------------------------------------

typedef __attribute__((ext_vector_type(4))) float v4f;
// Exact pointee type the async builtins expect: 'V4i' == GCC vector of 4 ints.
typedef int v4i_b __attribute__((vector_size(16)));

#define AS1 __attribute__((address_space(1)))
#define AS3 __attribute__((address_space(3)))

#define NOPE_DIM        512
#define ROPE_DIM        64
#define BUF_DIM         576
#define CHUNKS_PER_ROW  144            // 576 floats / 4 per b128 chunk
#define NOPE_CHUNKS     128            // first 128 chunks -> k_nope
#define ROWS_PER_BLOCK  16
#define BLOCK_THREADS   256            // 8 waves (wave32)
#define CHUNKS_PER_THREAD 9            // 16*144 / 256

#if defined(__AMDGCN__) && \
    __has_builtin(__builtin_amdgcn_global_load_async_to_lds_b128) && \
    __has_builtin(__builtin_amdgcn_global_store_async_from_lds_b128) && \
    __has_builtin(__builtin_amdgcn_s_wait_asynccnt)
#define USE_ASYNC_LDS 1
#else
#define USE_ASYNC_LDS 0
#endif

#if USE_ASYNC_LDS
// One 16-row tile: global -> LDS (async DMA) -> global. Each thread stores
// exactly the LDS slots it loaded, so the per-wave s_wait_asynccnt between
// the two phases is the only synchronization needed (no block barrier).
// GUARD=false is the hot path: no per-chunk EXEC masking at all.
template <bool GUARD>
__device__ __forceinline__ void tile_dma(
    const float* __restrict__ kv, const long long* __restrict__ loc,
    float* __restrict__ out, float* __restrict__ out_rope,
    v4f* stage, int row0, int tid, int n_loc)
{
    #pragma unroll
    for (int j = 0; j < CHUNKS_PER_THREAD; ++j) {
        const int c   = j * BLOCK_THREADS + tid;       // chunk within tile
        const int r   = c / CHUNKS_PER_ROW;
        const int k   = c - r * CHUNKS_PER_ROW;
        const int row = row0 + r;
        if (!GUARD || row < n_loc) {
            const long long slot = loc[row];           // 16 slots = 1 cacheline
            const float* src = kv + (size_t)slot * BUF_DIM + (size_t)(k * 4);
            __builtin_amdgcn_global_load_async_to_lds_b128(
                (AS1 v4i_b*)src, (AS3 v4i_b*)&stage[c],
                /*imm offset=*/0, /*cpol RT=*/0);
        }
    }
    __builtin_amdgcn_s_wait_asynccnt(0);   // our LDS slots are now written

    #pragma unroll
    for (int j = 0; j < CHUNKS_PER_THREAD; ++j) {
        const int c   = j * BLOCK_THREADS + tid;
        const int r   = c / CHUNKS_PER_ROW;
        const int k   = c - r * CHUNKS_PER_ROW;
        const int row = row0 + r;
        if (!GUARD || row < n_loc) {
            float* dst = (k < NOPE_CHUNKS)
                ? out      + (size_t)row * NOPE_DIM + (size_t)(k * 4)
                : out_rope + (size_t)row * ROPE_DIM + (size_t)((k - NOPE_CHUNKS) * 4);
            // TH=NT on the streaming output: don't evict reusable kv rows.
            __builtin_amdgcn_global_store_async_from_lds_b128(
                (AS1 v4i_b*)dst, (AS3 v4i_b*)&stage[c],
                /*imm offset=*/0, /*cpol NT=*/1);
        }
    }
    __builtin_amdgcn_s_wait_asynccnt(0);   // drain before wave retires
}
#endif

__global__ __launch_bounds__(BLOCK_THREADS) void gather_mla_rows(
    const float* __restrict__ kv, const long long* __restrict__ loc,
    float* __restrict__ out, int n_loc)
{
    const int tid  = threadIdx.x;
    const int row0 = blockIdx.x * ROWS_PER_BLOCK;
    float* const out_rope = out + (size_t)n_loc * NOPE_DIM;

#if USE_ASYNC_LDS
    __shared__ v4f stage[ROWS_PER_BLOCK * CHUNKS_PER_ROW];   // 36,864 B

    if (row0 + ROWS_PER_BLOCK <= n_loc) {
        tile_dma<false>(kv, loc, out, out_rope, stage, row0, tid, n_loc);
    } else {
        tile_dma<true >(kv, loc, out, out_rope, stage, row0, tid, n_loc);
    }
#else
    // Fallback: direct b128 gather through VGPRs with non-temporal stores.
    #pragma unroll
    for (int j = 0; j < CHUNKS_PER_THREAD; ++j) {
        const int c   = j * BLOCK_THREADS + tid;
        const int r   = c / CHUNKS_PER_ROW;
        const int k   = c - r * CHUNKS_PER_ROW;
        const int row = row0 + r;
        if (row < n_loc) {
            const long long slot = loc[row];
            const v4f v = *(const v4f*)(kv + (size_t)slot * BUF_DIM + (size_t)(k * 4));
            float* dst = (k < NOPE_CHUNKS)
                ? out      + (size_t)row * NOPE_DIM + (size_t)(k * 4)
                : out_rope + (size_t)row * ROPE_DIM + (size_t)((k - NOPE_CHUNKS) * 4);
            __builtin_nontemporal_store(v, (v4f*)dst);
        }
    }
#endif
}

extern "C" void kernel_launch(void* const* d_in, const int* in_sizes, int n_in,
                              void* d_out, int out_size, void* d_ws, size_t ws_size,
                              hipStream_t stream) {
    (void)n_in; (void)out_size; (void)d_ws; (void)ws_size;
    const float*     kv  = (const float*)d_in[0];
    const long long* loc = (const long long*)d_in[1];   // int64 per reference
    // d_in[2]/d_in[3] (zero-filled cache tensors) are unused; results go to d_out.
    float* out = (float*)d_out;

    const int n_loc  = in_sizes[1];                     // 65536
    const int blocks = (n_loc + ROWS_PER_BLOCK - 1) / ROWS_PER_BLOCK;  // 4096
    gather_mla_rows<<<blocks, BLOCK_THREADS, 0, stream>>>(kv, loc, out, n_loc);
}